// DynamicDeformConv2d_18348100289095
// MI455X (gfx1250) — compile-verified
//
#include <hip/hip_runtime.h>
#include <hip/hip_bf16.h>
#include <math.h>

// Problem constants (match reference setup_inputs)
#define B_    4
#define C_    64
#define H_    128
#define W_    128
#define HW_   (H_ * W_)
#define KTOT  576          // K = C_ * 9
#define FCH   576          // f-conv output channels
#define TILE_M 16          // pixels per workgroup (one WMMA M-tile)
#define WAVES  4
#define NT_PER_WAVE 9      // 36 N-tiles / 4 waves
#define THREADS (WAVES * 32)
#define LDS_STRIDE 580     // 576 + 4 pad -> bank-conflict-free (580 % 64 == 4)

typedef float v2f __attribute__((ext_vector_type(2)));
typedef float v8f __attribute__((ext_vector_type(8)));

#if defined(__HIP_DEVICE_COMPILE__) && defined(__gfx1250__) && \
    __has_builtin(__builtin_amdgcn_global_load_async_to_lds_b32)
#define HAVE_ASYNC_LDS 1
typedef __attribute__((address_space(1))) int g_i32;   // global int
typedef __attribute__((address_space(3))) int l_i32;   // LDS int
#else
#define HAVE_ASYNC_LDS 0
#endif

__device__ __forceinline__ void wait_asynccnt0() {
#if defined(__HIP_DEVICE_COMPILE__) && defined(__gfx1250__)
#if __has_builtin(__builtin_amdgcn_s_wait_asynccnt)
    __builtin_amdgcn_s_wait_asynccnt(0);
#else
    asm volatile("s_wait_asynccnt 0x0" ::: "memory");
#endif
#endif
}

// ---------------------------------------------------------------------------
// Kernel 0: transpose+pair-interleave f_conv_w (oc, ic, kh, kw) ->
//   wtp[k/2][oc][2], k = ic*9 + kh*3 + kw.  A single b64 load then yields
//   (B[k,oc], B[k+1,oc]) — exactly one WMMA B fragment per lane.
// ---------------------------------------------------------------------------
__global__ void k_weight_transpose(const float* __restrict__ fw,
                                   float* __restrict__ wtp) {
    int i = blockIdx.x * blockDim.x + threadIdx.x;      // over oc*576 (coalesced read)
    if (i < FCH * KTOT) {
        int oc = i / KTOT;
        int k  = i % KTOT;
        wtp[(size_t)(k >> 1) * (FCH * 2) + oc * 2 + (k & 1)] = fw[i];
    }
}

// ---------------------------------------------------------------------------
// Kernel 1: offset conv (18 ch) + sigmoid(mask conv) (9 ch) from feature_size
// ---------------------------------------------------------------------------
__global__ void k_conv_pm(const float* __restrict__ fs,
                          const float* __restrict__ pw, const float* __restrict__ pb,
                          const float* __restrict__ mw, const float* __restrict__ mb,
                          float* __restrict__ offb, float* __restrict__ mbuf) {
    int idx = blockIdx.x * blockDim.x + threadIdx.x;
    const int TOT = B_ * 27 * HW_;
    if (idx >= TOT) return;
    int hw = idx % HW_;
    int t  = idx / HW_;
    int ch = t % 27;
    int b  = t / 27;
    int h = hw / W_, w = hw % W_;

    const float* wgt;
    float acc;
    if (ch < 18) { wgt = pw + ch * KTOT;        acc = pb[ch]; }
    else         { wgt = mw + (ch - 18) * KTOT; acc = mb[ch - 18]; }

    const float* inb = fs + (size_t)b * C_ * HW_;
    for (int ic = 0; ic < C_; ++ic) {
        const float* ip = inb + ic * HW_;
        const float* wp = wgt + ic * 9;
        #pragma unroll
        for (int kh = 0; kh < 3; ++kh) {
            int hh = h + kh - 1;
            if (hh < 0 || hh >= H_) continue;
            #pragma unroll
            for (int kw = 0; kw < 3; ++kw) {
                int ww = w + kw - 1;
                if (ww < 0 || ww >= W_) continue;
                acc = fmaf(ip[hh * W_ + ww], wp[kh * 3 + kw], acc);
            }
        }
    }
    if (ch < 18) offb[((size_t)b * 18 + ch) * HW_ + hw] = acc;
    else         mbuf[((size_t)b * 9 + (ch - 18)) * HW_ + hw] =
                     1.0f / (1.0f + __expf(-acc));
}

// ---------------------------------------------------------------------------
// Kernel 2: fused WMMA implicit-GEMM f-conv + deformable sampling + einsum
// One workgroup: 16 pixels (fixed b,h; 16 consecutive w) x all 576 channels.
// ---------------------------------------------------------------------------
__global__ void __launch_bounds__(THREADS)
k_main(const float* __restrict__ fc,     // feature_context
       const float* __restrict__ xin,    // x
       const float* __restrict__ wtp,    // pair-interleaved transposed weight
       const float* __restrict__ fb,     // f_conv_b
       const float* __restrict__ offb,   // offsets (B,18,H,W)
       const float* __restrict__ mbuf,   // sigmoid masks (B,9,H,W)
       float* __restrict__ out) {
    __shared__ float ldsA[TILE_M * LDS_STRIDE];  // A patch, later reused for filt
    __shared__ int   sIdx[TILE_M * 9][4];        // gather indices per (pixel, tap)
    __shared__ float sWgt[TILE_M * 9][4];        // mask-folded bilinear weights

    const int tile = blockIdx.x;
    const int tw = tile & 7;                 // W_/TILE_M = 8
    const int h  = (tile >> 3) & (H_ - 1);
    const int b  = tile >> 10;               // 8 * 128 tiles per batch
    const int w0 = tw * TILE_M;

    const int tid   = threadIdx.x;
    const int wv    = tid >> 5;
    const int lane  = tid & 31;
    const int lm    = lane & 15;             // m (A) / n-in-tile (B,C,D)
    const int khalf = lane >> 4;             // K half-select for A/B fragments

    // ---- Phase A: stage 16x576 activation patch into LDS (zero padded) ----
    const float* fcb = fc + (size_t)b * C_ * HW_;
    const bool interior = (h >= 1) && (h <= H_ - 2) && (tw >= 1) && (tw <= 6);
#if HAVE_ASYNC_LDS
    if (interior) {
        // No padding needed: direct async global->LDS copies (ASYNCcnt),
        // overlapped with the sample-table phase below.
        for (int e = tid; e < TILE_M * KTOT; e += THREADS) {
            int m = e / KTOT, k = e % KTOT;
            int ic = k / 9, r = k % 9, kh = r / 3, kw = r % 3;
            const float* g = fcb + (size_t)ic * HW_
                           + (h + kh - 1) * W_ + (w0 + m + kw - 1);
            __builtin_amdgcn_global_load_async_to_lds_b32(
                (g_i32*)g, (l_i32*)&ldsA[m * LDS_STRIDE + k], 0, 0);
        }
    } else
#endif
    {
        for (int e = tid; e < TILE_M * KTOT; e += THREADS) {
            int m = e / KTOT, k = e % KTOT;
            int ic = k / 9, r = k % 9, kh = r / 3, kw = r % 3;
            int hh = h + kh - 1;
            int ww = w0 + m + kw - 1;
            float v = 0.0f;
            if (hh >= 0 && hh < H_ && ww >= 0 && ww < W_)
                v = fcb[(size_t)ic * HW_ + hh * W_ + ww];
            ldsA[m * LDS_STRIDE + k] = v;
        }
    }

    // ---- Phase A2: deformable-sample table, one entry per (pixel m, tap kk) --
    for (int e = tid; e < TILE_M * 9; e += THREADS) {
        int m  = e / 9, kk = e % 9;
        int kh = kk / 3, kw = kk % 3;
        int hp = h - (kh == 0 ? 1 : 0);      // source pixel row (shifted re-fold)
        int wp = w0 + m - (kw == 0 ? 1 : 0); // source pixel col
        int kx = (kh == 0) ? 2 : (kh - 1);   // sample index within 3x3
        int ky = (kw == 0) ? 2 : (kw - 1);
        int n  = kx * 3 + ky;

        int   ii[4] = {0, 0, 0, 0};
        float gg[4] = {0.f, 0.f, 0.f, 0.f};
        if (hp >= 0 && wp >= 0) {
            size_t pix = (size_t)hp * W_ + wp;
            float ox  = offb[((size_t)b * 18 + n) * HW_ + pix];
            float oy  = offb[((size_t)b * 18 + 9 + n) * HW_ + pix];
            float msk = mbuf[((size_t)b * 9 + n) * HW_ + pix];
            float px = 1.0f + (float)hp + (float)(kx - 1) + ox;   // padded coords
            float py = 1.0f + (float)wp + (float)(ky - 1) + oy;
            float fx = floorf(px), fy = floorf(py);
            const float HI = (float)(H_ + 1), WI = (float)(W_ + 1); // Hp-1, Wp-1
            float ltx = fminf(fmaxf(fx, 0.f), HI);
            float lty = fminf(fmaxf(fy, 0.f), WI);
            float rbx = fminf(fmaxf(fx + 1.f, 0.f), HI);
            float rby = fminf(fmaxf(fy + 1.f, 0.f), WI);
            float pxc = fminf(fmaxf(px, 0.f), HI);
            float pyc = fminf(fmaxf(py, 0.f), WI);
            float glt = (1.f + (ltx - pxc)) * (1.f + (lty - pyc));
            float grb = (1.f - (rbx - pxc)) * (1.f - (rby - pyc));
            float glb = (1.f + (ltx - pxc)) * (1.f - (rby - pyc));
            float grt = (1.f - (rbx - pxc)) * (1.f + (lty - pyc));
            int lxi = (int)ltx, lyi = (int)lty, rxi = (int)rbx, ryi = (int)rby;
            int ci[4][2] = {{lxi, lyi}, {rxi, ryi}, {lxi, ryi}, {rxi, lyi}};
            float cg[4]  = {glt, grb, glb, grt};
            #pragma unroll
            for (int q = 0; q < 4; ++q) {
                int ix = ci[q][0], iy = ci[q][1];
                bool valid = (ix >= 1 && ix <= H_ && iy >= 1 && iy <= W_);
                ii[q] = valid ? ((ix - 1) * W_ + (iy - 1)) : 0;
                gg[q] = valid ? cg[q] * msk : 0.0f;
            }
        }
        #pragma unroll
        for (int q = 0; q < 4; ++q) { sIdx[e][q] = ii[q]; sWgt[e][q] = gg[q]; }
    }
#if HAVE_ASYNC_LDS
    wait_asynccnt0();                         // async LDS copies complete
#endif
    __syncthreads();

    // ---- Phase B: implicit GEMM, fp32 WMMA 16x16x4, bias in accumulator ----
    v8f acc[NT_PER_WAVE];
    #pragma unroll
    for (int t = 0; t < NT_PER_WAVE; ++t) {
        int ch = (wv * NT_PER_WAVE + t) * 16 + lm;
        float bv = fb[ch];
        #pragma unroll
        for (int j = 0; j < 8; ++j) acc[t][j] = bv;
    }

    for (int k0 = 0; k0 < KTOT; k0 += 4) {
        int ka = k0 + khalf * 2;
        v2f afrag = *(const v2f*)&ldsA[lm * LDS_STRIDE + ka];  // ds_load_b64
        const float* wrow = wtp + (size_t)(ka >> 1) * (FCH * 2);
        #pragma unroll
        for (int t = 0; t < NT_PER_WAVE; ++t) {
            int ch = (wv * NT_PER_WAVE + t) * 16 + lm;
            v2f bfrag = *(const v2f*)&wrow[ch * 2];            // one b64 per frag
            acc[t] = __builtin_amdgcn_wmma_f32_16x16x4_f32(
                false, afrag, false, bfrag, (short)0, acc[t], false, false);
        }
    }

    // ---- Phase C: park filt tile (16 pixels x 576 ch) in LDS (reuse ldsA) ---
    __syncthreads();
    #pragma unroll
    for (int t = 0; t < NT_PER_WAVE; ++t) {
        int ch = (wv * NT_PER_WAVE + t) * 16 + lm;
        #pragma unroll
        for (int j = 0; j < 8; ++j) {
            int m = j + khalf * 8;                 // C/D layout: vgpr j -> M=j / j+8
            ldsA[m * LDS_STRIDE + ch] = acc[t][j];
        }
    }
    __syncthreads();

    // ---- Phase D: deformable gather + 9-tap einsum epilogue ----------------
    const float* xb = xin + (size_t)b * C_ * HW_;
    for (int p = tid; p < TILE_M * C_; p += THREADS) {
        int m = p & (TILE_M - 1);
        int c = p / TILE_M;
        const float* xc = xb + (size_t)c * HW_;
        float sum = 0.0f;
        #pragma unroll
        for (int kk = 0; kk < 9; ++kk) {
            int e = m * 9 + kk;
            float fv = ldsA[m * LDS_STRIDE + c * 9 + kk];
            float s = sWgt[e][0] * xc[sIdx[e][0]]
                    + sWgt[e][1] * xc[sIdx[e][1]]
                    + sWgt[e][2] * xc[sIdx[e][2]]
                    + sWgt[e][3] * xc[sIdx[e][3]];
            sum = fmaf(fv, s, sum);
        }
        out[(((size_t)b * C_ + c) * H_ + h) * W_ + w0 + m] = sum;
    }
}

// ---------------------------------------------------------------------------
extern "C" void kernel_launch(void* const* d_in, const int* in_sizes, int n_in,
                              void* d_out, int out_size, void* d_ws, size_t ws_size,
                              hipStream_t stream) {
    const float* feature_size    = (const float*)d_in[0];
    const float* feature_context = (const float*)d_in[1];
    const float* x               = (const float*)d_in[2];
    const float* p_conv_w        = (const float*)d_in[3];
    const float* p_conv_b        = (const float*)d_in[4];
    const float* f_conv_w        = (const float*)d_in[5];
    const float* f_conv_b        = (const float*)d_in[6];
    const float* m_conv_w        = (const float*)d_in[7];
    const float* m_conv_b        = (const float*)d_in[8];
    float* out = (float*)d_out;

    float* ws   = (float*)d_ws;
    float* offb = ws;                               // B*18*HW
    float* mbuf = offb + (size_t)B_ * 18 * HW_;     // B*9*HW
    float* wtp  = mbuf + (size_t)B_ * 9 * HW_;      // 576*576 (pair-interleaved)

    // 0) weight transpose + pair interleave
    {
        int tot = FCH * KTOT;
        k_weight_transpose<<<(tot + 255) / 256, 256, 0, stream>>>(f_conv_w, wtp);
    }
    // 1) offset + mask convs
    {
        int tot = B_ * 27 * HW_;
        k_conv_pm<<<(tot + 255) / 256, 256, 0, stream>>>(
            feature_size, p_conv_w, p_conv_b, m_conv_w, m_conv_b, offb, mbuf);
    }
    // 2) fused WMMA conv + sampling + einsum
    {
        int tiles = B_ * H_ * (W_ / TILE_M);        // 4096
        k_main<<<tiles, THREADS, 0, stream>>>(
            feature_context, x, wtp, f_conv_b, offb, mbuf, out);
    }
}